// LOUPEDynamicPolicy_76570676953369
// MI455X (gfx1250) — compile-verified
//
#include <hip/hip_runtime.h>
#include <hip/hip_bf16.h>
#include <stdint.h>

// Problem constants (from reference)
#define B_ 4
#define C_ 8
#define T_ 12
#define H_ 256
#define W_ 256
#define SLOPE_ 10.0f
#define BUDGET_ 43.0f

typedef float v4f __attribute__((ext_vector_type(4)));
typedef float v2f __attribute__((ext_vector_type(2)));

#define USE_ASYNC_LDS 1

// ---------------------------------------------------------------------------
// Stage 1: per-(b,t) policy math on W=256 columns.  48 blocks x 256 threads.
// Writes om[b,t,w] = m + hard  and  normed_last[b,w] = normed[b,T-1,w] to ws.
// ---------------------------------------------------------------------------
__global__ __launch_bounds__(256) void policy_kernel(const float* __restrict__ mask,
                                                     const float* __restrict__ sampler,
                                                     const float* __restrict__ u,
                                                     float* __restrict__ om,
                                                     float* __restrict__ normed_last) {
    const int bt = blockIdx.x;          // 0 .. B*T-1
    const int b  = bt / T_;
    const int t  = bt % T_;
    const int w  = threadIdx.x;         // 0 .. 255 (== W)

    __shared__ float red[W_];

    // m = mask[b,0,0,0,w,0]
    const float m    = mask[(size_t)b * (T_ * H_ * W_) + w];
    const float freev = 1.0f - m;

    // softplus(SLOPE * sampler)/SLOPE  (logaddexp(x,0) = max(x,0)+log1p(exp(-|x|)))
    const float x  = SLOPE_ * sampler[t * W_ + w];
    const float sp = (fmaxf(x, 0.0f) + log1pf(expf(-fabsf(x)))) * (1.0f / SLOPE_);

    // denom = max_w(free * sp)
    red[w] = freev * sp;
    __syncthreads();
    for (int s = W_ / 2; s > 0; s >>= 1) {
        if (w < s) red[w] = fmaxf(red[w], red[w + s]);
        __syncthreads();
    }
    const float denom = red[0];
    __syncthreads();

    // n_free = sum_w(free)
    red[w] = freev;
    __syncthreads();
    for (int s = W_ / 2; s > 0; s >>= 1) {
        if (w < s) red[w] += red[w + s];
        __syncthreads();
    }
    const float n_free = red[0];
    __syncthreads();

    const float prob = sp / denom;
    const float mp   = prob * freev;

    // xbar = sum_w(mp) / n_free
    red[w] = mp;
    __syncthreads();
    for (int s = W_ / 2; s > 0; s >>= 1) {
        if (w < s) red[w] += red[w + s];
        __syncthreads();
    }
    const float xbar = red[0] / n_free;

    const float sparsity = BUDGET_ / n_free;
    const float r    = sparsity / xbar;
    const float beta = (1.0f - sparsity) / (1.0f - xbar);

    float normed = (r <= 1.0f) ? (mp * r) : (1.0f - (1.0f - mp) * beta);
    normed *= freev;

    const float uu   = u[(size_t)bt * W_ + w];
    const float hard = (normed > uu) ? 1.0f : 0.0f;   // forward value of acq

    om[(size_t)bt * W_ + w] = m + hard;               // out_mask row value (0 or 1)
    if (t == T_ - 1) normed_last[b * W_ + w] = normed;
}

// ---------------------------------------------------------------------------
// Stage 2a: masked_kspace = out_mask ? kspace : +0.0
// Layout (B,C,T,H,W,2): one block owns 1/4 of a (b,c,t) plane.
// om row (1KB) staged into LDS via CDNA5 async global->LDS engine.
// ---------------------------------------------------------------------------
#define KS_PLANE_F4 (H_ * W_ * 2 / 4)   // 32768 float4 per (b,c,t)
#define KS_SPLIT 4

__global__ __launch_bounds__(256) void mask_kspace_kernel(const v4f* __restrict__ ks4,
                                                          const float* __restrict__ om,
                                                          v4f* __restrict__ out4) {
    const int bid   = blockIdx.x;             // B*C*T*KS_SPLIT = 1536
    const int plane = bid >> 2;               // (b*C + c)*T + t
    const int part  = bid & (KS_SPLIT - 1);
    const int t     = plane % T_;
    const int b     = (plane / T_) / C_;

    __shared__ __align__(16) float s_om[W_];
    const float* omrow = om + ((size_t)b * T_ + t) * W_;

#if USE_ASYNC_LDS
    if (threadIdx.x < W_ / 4) {               // 64 lanes x 16B = 1KB row
        unsigned long long ga = (unsigned long long)(const void*)(omrow + threadIdx.x * 4);
        unsigned int       la = (unsigned int)(unsigned long long)(const void*)(&s_om[threadIdx.x * 4]);
        asm volatile("global_load_async_to_lds_b128 %0, %1, off"
                     :: "v"(la), "v"(ga) : "memory");
        asm volatile("s_wait_asynccnt 0" ::: "memory");
    }
    __syncthreads();
#else
    s_om[threadIdx.x] = omrow[threadIdx.x];
    __syncthreads();
#endif

    const v2f* s_om2 = (const v2f*)s_om;      // pair (w0, w0+1) per float4 of kspace
    const size_t base = (size_t)plane * KS_PLANE_F4;
    const int per = KS_PLANE_F4 / KS_SPLIT;   // 8192
    const int end = (part + 1) * per;

#pragma unroll 8
    for (int i = part * per + threadIdx.x; i < end; i += 256) {
        v4f k = __builtin_nontemporal_load(&ks4[base + i]);
        v2f o = s_om2[i & (W_ / 2 - 1)];
        v4f rr;
        rr.x = (o.x != 0.0f) ? k.x : 0.0f;
        rr.y = (o.x != 0.0f) ? k.y : 0.0f;
        rr.z = (o.y != 0.0f) ? k.z : 0.0f;
        rr.w = (o.y != 0.0f) ? k.w : 0.0f;
        __builtin_nontemporal_store(rr, &out4[base + i]);
    }
}

// ---------------------------------------------------------------------------
// Stage 2b: out_mask (B,1,T,H,W,1) = broadcast om[b,t,w] over H.
// ---------------------------------------------------------------------------
#define OM_PLANE_F4 (H_ * W_ / 4)   // 16384 float4 per (b,t)
#define OM_SPLIT 16

__global__ __launch_bounds__(256) void out_mask_kernel(const float* __restrict__ om,
                                                       v4f* __restrict__ out4) {
    const int bid   = blockIdx.x;                   // B*T*OM_SPLIT = 768
    const int plane = bid / OM_SPLIT;               // b*T + t
    const int part  = bid % OM_SPLIT;

    __shared__ __align__(16) float s_om[W_];
    s_om[threadIdx.x] = om[(size_t)plane * W_ + threadIdx.x];
    __syncthreads();

    const v4f* s4 = (const v4f*)s_om;               // 64 quads of w
    const size_t base = (size_t)plane * OM_PLANE_F4;
    const int per = OM_PLANE_F4 / OM_SPLIT;         // 1024
    const int end = (part + 1) * per;

#pragma unroll 4
    for (int i = part * per + threadIdx.x; i < end; i += 256) {
        __builtin_nontemporal_store(s4[i & 63], &out4[base + i]);
    }
}

// ---------------------------------------------------------------------------
// Stage 2c: final_prob (B,1,H,W,1) = broadcast normed_last[b,w] over H.
// 256 blocks x 256 threads, one float4 each (65536 total).
// ---------------------------------------------------------------------------
__global__ __launch_bounds__(256) void final_prob_kernel(const float* __restrict__ nl,
                                                         v4f* __restrict__ out4) {
    const int g = blockIdx.x * 256 + threadIdx.x;   // 0 .. 65535
    const int b = g >> 14;                          // 16384 float4 per b
    const int q = g & 63;                           // w-quad
    const v4f* nl4 = (const v4f*)nl;
    __builtin_nontemporal_store(nl4[b * 64 + q], &out4[g]);
}

// ---------------------------------------------------------------------------
extern "C" void kernel_launch(void* const* d_in, const int* in_sizes, int n_in,
                              void* d_out, int out_size, void* d_ws, size_t ws_size,
                              hipStream_t stream) {
    const float* mask    = (const float*)d_in[0];
    const float* kspace  = (const float*)d_in[1];
    const float* sampler = (const float*)d_in[2];
    const float* u       = (const float*)d_in[3];
    float* out = (float*)d_out;
    float* ws  = (float*)d_ws;

    float* om = ws;                       // B*T*W floats = 12288
    float* nl = ws + (size_t)B_ * T_ * W_; // B*W floats   = 1024

    policy_kernel<<<B_ * T_, 256, 0, stream>>>(mask, sampler, u, om, nl);

    mask_kspace_kernel<<<B_ * C_ * T_ * KS_SPLIT, 256, 0, stream>>>(
        (const v4f*)kspace, om, (v4f*)out);

    const size_t ks_elems = (size_t)B_ * C_ * T_ * H_ * W_ * 2;  // 50331648
    out_mask_kernel<<<B_ * T_ * OM_SPLIT, 256, 0, stream>>>(
        om, (v4f*)(out + ks_elems));

    const size_t om_elems = (size_t)B_ * T_ * H_ * W_;           // 3145728
    final_prob_kernel<<<256, 256, 0, stream>>>(
        nl, (v4f*)(out + ks_elems + om_elems));
}